// GINELayer_48009144434788
// MI455X (gfx1250) — compile-verified
//
#include <hip/hip_runtime.h>
#include <hip/hip_bf16.h>

#define D      128
#define ED     32
#define BN_EPS 1e-5f
#define WSTR   (2 * D + 32)   // pair-row stride in floats (+32 = bank-phase pad)

typedef float v2f __attribute__((ext_vector_type(2)));
typedef float v8f __attribute__((ext_vector_type(8)));

static __device__ __forceinline__ v8f wmma_f32_4(v2f a, v2f b, v8f c) {
    // D = A(16x4 f32) * B(4x16 f32) + C(16x16 f32)
    return __builtin_amdgcn_wmma_f32_16x16x4_f32(
        /*neg_a=*/false, a, /*neg_b=*/false, b,
        /*c_mod=*/(short)0, c, /*reuse_a=*/false, /*reuse_b=*/false);
}

// ---------------------------------------------------------------- utilities
__global__ void __launch_bounds__(256) k_zero(float* p, int n) {
    int i = blockIdx.x * 256 + threadIdx.x;
    if (i < n) p[i] = 0.0f;
}

__global__ void __launch_bounds__(256) k_scale(const float* __restrict__ x,
                                               const float* __restrict__ eps,
                                               float* __restrict__ h, int total) {
    int i = blockIdx.x * 256 + threadIdx.x;
    if (i < total) h[i] = (1.0f + eps[0]) * x[i];
}

// ------------------------------------------------- edge GEMM + gather/scatter
// Each wave: 16 edges. A = edge_attr[16x32] (fp32 WMMA, 8 K-steps),
// B = We staged pair-interleaved in LDS (one ds_load_b64 per fragment).
// All 8 output tiles kept in registers, then row-wise epilogue:
// per source row, 8 clause-grouped gathers + 8 clause-grouped atomics.
__global__ void __launch_bounds__(256) k_edge(const float* __restrict__ ea,
                                              const int* __restrict__ ei,
                                              const float* __restrict__ x,
                                              const float* __restrict__ We,
                                              const float* __restrict__ be,
                                              float* __restrict__ h,
                                              int E, int ntiles) {
    __shared__ __align__(16) float Ws[(ED / 2) * WSTR];   // 18 KB, pair-interleaved
    __shared__ __align__(16) int sSrc[8][16];
    __shared__ __align__(16) int sDst[8][16];

    const int tid = threadIdx.x;
    // stage We[k][n] -> Ws[(k>>1)*WSTR + 2n + (k&1)]
    for (int i = tid; i < ED * D; i += 256) {
        int k = i >> 7, n = i & (D - 1);
        Ws[(k >> 1) * WSTR + 2 * n + (k & 1)] = We[i];
    }

    const int wave = tid >> 5;
    const int lane = tid & 31;
    const int wid  = blockIdx.x * 8 + wave;
    const bool active = (wid < ntiles);
    const int  e0 = wid * 16;

    if (active && lane < 16) {
        sSrc[wave][lane] = ei[e0 + lane];            // row 0 of edge_index = src
        sDst[wave][lane] = ei[E + e0 + lane];        // row 1 of edge_index = dst
    }
    __syncthreads();
    if (!active) return;                             // wave-uniform branch

    const int m     = lane & 15;                     // M (and N) position
    const int khalf = (lane >> 4) * 2;               // lanes 16-31 hold K=2,3
    const int phase = lane >> 4;                     // pair-row phase

    // preload all 8 A fragments (K = 0..31 in steps of 4)
    v2f aF[8];
    const float* arow = ea + (size_t)(e0 + m) * ED;
    #pragma unroll
    for (int kt = 0; kt < 8; ++kt) {
        int k0 = kt * 4 + khalf;
        aF[kt].x = arow[k0];
        aF[kt].y = arow[k0 + 1];
    }

    // compute all 8 N-tiles (16x128 output block held in 64 VGPRs)
    v8f acc[8];
    #pragma unroll
    for (int t = 0; t < 8; ++t) {
        acc[t] = (v8f){};
        #pragma unroll
        for (int kt = 0; kt < 8; ++kt) {
            int p = kt * 2 + phase;                  // K-pair index
            v2f b = *(const v2f*)&Ws[p * WSTR + 2 * (t * 16 + m)];
            acc[t] = wmma_f32_4(aF[kt], b, acc[t]);
        }
    }

    // preload biases for this lane's 8 columns
    float bias[8];
    #pragma unroll
    for (int t = 0; t < 8; ++t) bias[t] = be[t * 16 + m];

    // preload this lane's 8 src/dst row offsets (rows rbase..rbase+7)
    const int rbase = (lane < 16) ? 0 : 8;
    size_t soff[8], doff[8];
    #pragma unroll
    for (int v = 0; v < 8; ++v) {
        soff[v] = (size_t)sSrc[wave][rbase + v] * D;
        doff[v] = (size_t)sDst[wave][rbase + v] * D;
    }

    // row-wise epilogue: relu(acc+bias) + gather x[src] -> atomicAdd h[dst]
    #pragma unroll
    for (int v = 0; v < 8; ++v) {
        const float* xs = x + soff[v];
        float*       hd = h + doff[v];
        float msg[8];
        #pragma unroll
        for (int t = 0; t < 8; ++t) {                // 8 gathers, clause-grouped
            float val = acc[t][v] + bias[t];
            val = val > 0.0f ? val : 0.0f;           // ReLU
            msg[t] = val + xs[t * 16 + m];
        }
        #pragma unroll
        for (int t = 0; t < 8; ++t) {                // 8 atomics, clause-grouped
            unsafeAtomicAdd(&hd[t * 16 + m], msg[t]);
        }
    }
}

// scalar fallback for E % 16 tail edges
__global__ void __launch_bounds__(256) k_edge_rem(const float* __restrict__ ea,
                                                  const int* __restrict__ ei,
                                                  const float* __restrict__ x,
                                                  const float* __restrict__ We,
                                                  const float* __restrict__ be,
                                                  float* __restrict__ h,
                                                  int E, int estart) {
    int idx = blockIdx.x * 256 + threadIdx.x;
    int rem = E - estart;
    if (idx >= rem * D) return;
    int e = estart + idx / D;
    int c = idx & (D - 1);
    float acc = be[c];
    for (int k = 0; k < ED; ++k) acc += ea[(size_t)e * ED + k] * We[k * D + c];
    acc = acc > 0.0f ? acc : 0.0f;
    int s = ei[e], d = ei[E + e];
    unsafeAtomicAdd(&h[(size_t)d * D + c], acc + x[(size_t)s * D + c]);
}

// ------------------------------------------------------ node GEMM (X @ W + b)
// Block = 8 waves; wave w computes rows [blk*128 + w*16, +16) x all 128 cols.
// Full W staged pair-interleaved in LDS (72 KB); 32 K-steps x 8 N-tiles of
// fp32 WMMA per wave, each B fragment a single aligned ds_load_b64.
__global__ void __launch_bounds__(256) k_gemm(const float* __restrict__ X,
                                              const float* __restrict__ W,
                                              const float* __restrict__ bias,
                                              float* __restrict__ Y, int Nrows) {
    __shared__ __align__(16) float Ws[(D / 2) * WSTR];    // 72 KB
    const int tid = threadIdx.x;
    for (int i = tid; i < D * D; i += 256) {
        int k = i >> 7, n = i & (D - 1);
        Ws[(k >> 1) * WSTR + 2 * n + (k & 1)] = W[i];
    }
    __syncthreads();

    const int wave  = tid >> 5;
    const int lane  = tid & 31;
    const int m     = lane & 15;
    const int khalf = (lane >> 4) * 2;
    const int phase = lane >> 4;
    const int rowBase = blockIdx.x * 128 + wave * 16;

    int grow = rowBase + m;
    if (grow >= Nrows) grow = Nrows - 1;        // clamp (stores are guarded)
    const float* xr = X + (size_t)grow * D;

    v8f acc[8];
    #pragma unroll
    for (int t = 0; t < 8; ++t) acc[t] = (v8f){};

    for (int kt = 0; kt < 32; ++kt) {
        int k0 = kt * 4 + khalf;
        v2f a;
        a.x = xr[k0];
        a.y = xr[k0 + 1];
        int p = kt * 2 + phase;
        #pragma unroll
        for (int t = 0; t < 8; ++t) {
            v2f b = *(const v2f*)&Ws[p * WSTR + 2 * (t * 16 + m)];
            acc[t] = wmma_f32_4(a, b, acc[t]);
        }
    }

    #pragma unroll
    for (int t = 0; t < 8; ++t) {
        int   col = t * 16 + m;
        float bv  = bias[col];
        #pragma unroll
        for (int v = 0; v < 8; ++v) {
            int r = rowBase + ((lane < 16) ? v : v + 8);
            if (r < Nrows) Y[(size_t)r * D + col] = acc[t][v] + bv;
        }
    }
}

// --------------------------------------------- per-column batch stats (sum, sumsq)
__global__ void __launch_bounds__(256) k_colstats(const float* __restrict__ Y,
                                                  float* __restrict__ stats,
                                                  int Nrows) {
    __shared__ float red[512];
    const int tid  = threadIdx.x;
    const int c    = tid & (D - 1);
    const int half = tid >> 7;                 // 0 or 1
    float s = 0.0f, sq = 0.0f;
    for (int r = blockIdx.x * 2 + half; r < Nrows; r += gridDim.x * 2) {
        float v = Y[(size_t)r * D + c];
        s += v;
        sq += v * v;
    }
    red[tid] = s;
    red[256 + tid] = sq;
    __syncthreads();
    if (tid < 128) {
        unsafeAtomicAdd(&stats[c],     red[tid] + red[tid + 128]);
        unsafeAtomicAdd(&stats[D + c], red[256 + tid] + red[256 + tid + 128]);
    }
}

// -------------------------------------------------- BN (train stats) + ReLU
__global__ void __launch_bounds__(256) k_bnrelu(const float* __restrict__ Y,
                                                const float* __restrict__ stats,
                                                const float* __restrict__ g,
                                                const float* __restrict__ beta,
                                                float* __restrict__ Z, int Nrows) {
    int idx = blockIdx.x * 256 + threadIdx.x;
    int total = Nrows * D;
    if (idx >= total) return;
    int c = idx & (D - 1);
    float invN = 1.0f / (float)Nrows;
    float mean = stats[c] * invN;
    float var  = stats[D + c] * invN - mean * mean;
    float rs   = rsqrtf(var + BN_EPS);
    float v    = (Y[idx] - mean) * rs * g[c] + beta[c];
    Z[idx] = v > 0.0f ? v : 0.0f;
}

// ---------------------------------------------------------------- launcher
extern "C" void kernel_launch(void* const* d_in, const int* in_sizes, int n_in,
                              void* d_out, int out_size, void* d_ws, size_t ws_size,
                              hipStream_t stream) {
    const float* x     = (const float*)d_in[0];
    const int*   ei    = (const int*)  d_in[1];
    const float* ea    = (const float*)d_in[2];
    const float* eps   = (const float*)d_in[3];
    const float* We    = (const float*)d_in[4];
    const float* be    = (const float*)d_in[5];
    const float* W1    = (const float*)d_in[6];
    const float* b1    = (const float*)d_in[7];
    const float* g1    = (const float*)d_in[8];
    const float* beta1 = (const float*)d_in[9];
    const float* W2    = (const float*)d_in[10];
    const float* b2    = (const float*)d_in[11];
    const float* g2    = (const float*)d_in[12];
    const float* beta2 = (const float*)d_in[13];

    const int N = in_sizes[0] / D;      // 50000 nodes
    const int E = in_sizes[1] / 2;      // 800000 edges
    const size_t ND = (size_t)N * D;

    float* h      = (float*)d_ws;       // [N, D] aggregation / intermediate
    float* y      = h + ND;             // [N, D] GEMM output
    float* stats1 = y + ND;             // [2, D]
    float* stats2 = stats1 + 2 * D;     // [2, D]
    float* out    = (float*)d_out;

    const int elemBlocks = (int)((ND + 255) / 256);

    // zero both stats buffers (4*D floats)
    k_zero<<<(4 * D + 255) / 256, 256, 0, stream>>>(stats1, 4 * D);

    // h = (1 + eps) * x
    k_scale<<<elemBlocks, 256, 0, stream>>>(x, eps, h, (int)ND);

    // edge embed (WMMA) + gather + scatter-add into h
    const int ntiles = E / 16;
    if (ntiles > 0)
        k_edge<<<(ntiles + 7) / 8, 256, 0, stream>>>(ea, ei, x, We, be, h, E, ntiles);
    const int rem = E - ntiles * 16;
    if (rem > 0)
        k_edge_rem<<<(rem * D + 255) / 256, 256, 0, stream>>>(ea, ei, x, We, be, h,
                                                              E, ntiles * 16);

    // layer 1: y = h @ W1 + b1 ; stats ; h = relu(bn(y))
    k_gemm<<<(N + 127) / 128, 256, 0, stream>>>(h, W1, b1, y, N);
    k_colstats<<<256, 256, 0, stream>>>(y, stats1, N);
    k_bnrelu<<<elemBlocks, 256, 0, stream>>>(y, stats1, g1, beta1, h, N);

    // layer 2: y = h @ W2 + b2 ; stats ; out = relu(bn(y))
    k_gemm<<<(N + 127) / 128, 256, 0, stream>>>(h, W2, b2, y, N);
    k_colstats<<<256, 256, 0, stream>>>(y, stats2, N);
    k_bnrelu<<<elemBlocks, 256, 0, stream>>>(y, stats2, g2, beta2, out, N);
}